// GCN_Decoder_89404039233749
// MI455X (gfx1250) — compile-verified
//
#include <hip/hip_runtime.h>
#include <hip/hip_bf16.h>
#include <math.h>
#include <stdint.h>

// GCN 2-layer decoder for MI455X (gfx1250, wave32).
// - Dense GEMMs: fp32 WMMA (V_WMMA_F32_16X16X4_F32), A-strip staged in LDS
//   via CDNA5 async global->LDS copies (global_load_async_to_lds_b128 +
//   s_wait_asynccnt), B fragments from L2-resident weights.
// - Graph aggregation: L2-resident gathers + f32 atomic scatters (the
//   roofline bottleneck: ~1.3 GB of mostly-L2 traffic vs 6.6 GF of GEMM).

typedef float v2f __attribute__((ext_vector_type(2)));
typedef float v8f __attribute__((ext_vector_type(8)));

#define WAVE_SIZE 32
#define GEMM_BLOCK_THREADS 128   // 4 waves per block
#define MAX_K 256                // largest K we stage in LDS (16 rows x 256 f32 = 16KB)

// ---------------------------------------------------------------------------
// Degree / normalization
// ---------------------------------------------------------------------------

__global__ void init_deg_kernel(float* __restrict__ deg, int n) {
    int i = blockIdx.x * blockDim.x + threadIdx.x;
    if (i < n) deg[i] = 1.0f;  // self-loop contributes 1 to every degree
}

__global__ void deg_accum_kernel(const long long* __restrict__ dst,
                                 float* __restrict__ deg, int e) {
    int i = blockIdx.x * blockDim.x + threadIdx.x;
    if (i < e) atomicAdd(&deg[(int)dst[i]], 1.0f);
}

__global__ void dinv_kernel(float* __restrict__ deg, int n) {
    int i = blockIdx.x * blockDim.x + threadIdx.x;
    if (i < n) {
        float d = deg[i];
        deg[i] = (d > 0.0f) ? rsqrtf(d) : 0.0f;  // in place: deg -> dinv
    }
}

// ---------------------------------------------------------------------------
// fp32 WMMA GEMM: C[M,N] = A[M,K] * B[K,N]
//
// Grid: M/16 blocks, 128 threads (4 waves). Each block owns a 16-row strip.
// The strip (16 x K floats, contiguous full rows) is staged into LDS once via
// async global->LDS b128 copies. Each wave computes 16 x (NACC*16) of C, so
// one A fragment feeds NACC WMMAs per K-step of 4.
//
// Fragment layouts (ISA 7.12.2, 32-bit, wave32):
//   A (16x4): lanes 0-15 hold M=lane, K={k,k+1}; lanes 16-31 K={k+2,k+3}.
//   B (4x16): VGPR0 lanes 0-15 = row k, lanes 16-31 = row k+2; VGPR1 = +1.
//   C (16x16): VGPR v, lanes 0-15 -> M=v; lanes 16-31 -> M=v+8; N=lane&15.
// ---------------------------------------------------------------------------

template <int NACC>
__global__ __launch_bounds__(GEMM_BLOCK_THREADS)
void gemm_wmma_f32_lds_kernel(const float* __restrict__ A,
                              const float* __restrict__ B,
                              float* __restrict__ C,
                              int M, int N, int K) {
    __shared__ float As[16 * MAX_K];

    const int tid  = threadIdx.x;
    const int lane = tid & (WAVE_SIZE - 1);
    const int wave = tid / WAVE_SIZE;
    const int row0 = blockIdx.x * 16;

    // --- stage the 16xK A strip into LDS (contiguous region, b128/lane) ----
    const float* astrip = A + (size_t)row0 * K;
    const int strip_elems = 16 * K;              // multiple of 512 for K in {128,256}
    for (int e = tid * 4; e < strip_elems; e += GEMM_BLOCK_THREADS * 4) {
        unsigned lds = (unsigned)(uintptr_t)(&As[e]);   // low 32 bits = LDS offset
        const float* g = astrip + e;
        asm volatile("global_load_async_to_lds_b128 %0, %1, off"
                     :: "v"(lds), "v"(g) : "memory");
    }
    asm volatile("s_wait_asynccnt 0x0" ::: "memory");
    __syncthreads();

    const int m16   = lane & 15;                 // A row / B,C col within tile
    const int khalf = (lane >> 4) << 1;          // 0 (lanes 0-15) or 2 (16-31)
    const int col0  = wave * (NACC * 16);        // wave's first output column

    v8f zero = {};
    v8f acc[NACC];
    #pragma unroll
    for (int j = 0; j < NACC; ++j) acc[j] = zero;

    const float* as_row = &As[m16 * K];

    #pragma unroll 2
    for (int k = 0; k < K; k += 4) {
        v2f a;                                   // ds_load_b64 (8B aligned)
        a.x = as_row[k + khalf];
        a.y = as_row[k + khalf + 1];
        #pragma unroll
        for (int j = 0; j < NACC; ++j) {
            const float* bptr = B + (size_t)(k + khalf) * N + col0 + j * 16 + m16;
            v2f b;
            b.x = bptr[0];
            b.y = bptr[(size_t)N];
            acc[j] = __builtin_amdgcn_wmma_f32_16x16x4_f32(
                /*neg_a=*/false, a, /*neg_b=*/false, b,
                /*c_mod=*/(short)0, acc[j], /*reuse_a=*/false, /*reuse_b=*/false);
        }
    }

    const int r0 = row0 + ((lane >> 4) << 3);    // +8 for upper half-wave
    #pragma unroll
    for (int j = 0; j < NACC; ++j) {
        float* cdst = C + (size_t)r0 * N + col0 + j * 16 + m16;
        #pragma unroll
        for (int v = 0; v < 8; ++v) cdst[(size_t)v * N] = acc[j][v];
    }
}

// ---------------------------------------------------------------------------
// Aggregation
// ---------------------------------------------------------------------------

// agg[i,:] = h[i,:] * dinv[i]^2 (+ bias): the self-loop term, which also
// initializes the accumulation buffer (d_ws / d_out are poisoned by harness).
__global__ void self_init_kernel(const float* __restrict__ h,
                                 const float* __restrict__ dinv,
                                 const float* __restrict__ bias,  // may be null
                                 float* __restrict__ agg,
                                 int n, int f) {
    long long idx = (long long)blockIdx.x * blockDim.x + threadIdx.x;
    long long total = (long long)n * f;
    if (idx >= total) return;
    int node = (int)(idx / f);
    int feat = (int)(idx - (long long)node * f);
    float di = dinv[node];
    float v = h[idx] * di * di;
    if (bias) v += bias[feat];
    agg[idx] = v;
}

// One wave per edge: agg[dst,:] += h[src,:] * (dinv[src]*dinv[dst]).
// Lanes stride the feature dimension -> coalesced gathers, coalesced atomics.
// h (51 MB / 26 MB) is L2-resident on MI455X (192 MB L2).
__global__ void edge_scatter_kernel(const float* __restrict__ h,
                                    const long long* __restrict__ src,
                                    const long long* __restrict__ dst,
                                    const float* __restrict__ dinv,
                                    float* __restrict__ agg,
                                    int e, int f) {
    int edge = (int)((blockIdx.x * (long long)blockDim.x + threadIdx.x) / WAVE_SIZE);
    int lane = threadIdx.x & (WAVE_SIZE - 1);
    if (edge >= e) return;
    int s = (int)src[edge];
    int d = (int)dst[edge];
    float nrm = dinv[s] * dinv[d];
    const float* hs = h + (size_t)s * f;
    float* ad = agg + (size_t)d * f;
    for (int i = lane; i < f; i += WAVE_SIZE) {
        atomicAdd(&ad[i], hs[i] * nrm);
    }
}

// z = relu(agg + b), in place.
__global__ void relu_bias_kernel(float* __restrict__ agg,
                                 const float* __restrict__ bias,
                                 int n, int f) {
    long long idx = (long long)blockIdx.x * blockDim.x + threadIdx.x;
    long long total = (long long)n * f;
    if (idx >= total) return;
    int feat = (int)(idx % f);
    float v = agg[idx] + bias[feat];
    agg[idx] = v > 0.0f ? v : 0.0f;
}

// ---------------------------------------------------------------------------
// Host-side orchestration (all on `stream`, graph-capture safe)
// ---------------------------------------------------------------------------

extern "C" void kernel_launch(void* const* d_in, const int* in_sizes, int n_in,
                              void* d_out, int out_size, void* d_ws, size_t ws_size,
                              hipStream_t stream) {
    const float*      x  = (const float*)d_in[0];
    const long long*  ei = (const long long*)d_in[1];   // int64 edge_index [2,E]
    const float*      W1 = (const float*)d_in[2];
    const float*      b1 = (const float*)d_in[3];
    const float*      W2 = (const float*)d_in[4];
    const float*      b2 = (const float*)d_in[5];
    float*            out = (float*)d_out;

    const int F_IN  = 128;
    const int F_HID = in_sizes[3];          // 256 (len of b1)
    const int F_OUT = in_sizes[5];          // 128 (len of b2)
    const int N     = in_sizes[0] / F_IN;   // 50000
    const int E     = in_sizes[1] / 2;      // 800000

    const long long* src = ei;              // edge_index[0]
    const long long* dst = ei + E;          // edge_index[1]

    // Workspace layout (fp32): dinv | h1 | agg1(z1) | h2
    float* ws   = (float*)d_ws;
    size_t off  = 0;
    float* dinv = ws + off; off += ((size_t)N + 255) & ~(size_t)255;
    float* h1   = ws + off; off += (size_t)N * F_HID;
    float* agg1 = ws + off; off += (size_t)N * F_HID;
    float* h2   = ws + off;

    const int TB = 256;

    // --- degree / dinv -----------------------------------------------------
    init_deg_kernel<<<(N + TB - 1) / TB, TB, 0, stream>>>(dinv, N);
    deg_accum_kernel<<<(E + TB - 1) / TB, TB, 0, stream>>>(dst, dinv, E);
    dinv_kernel<<<(N + TB - 1) / TB, TB, 0, stream>>>(dinv, N);

    // --- layer 1: h1 = x @ W1  (N x 128) @ (128 x 256) ---------------------
    // 4 waves/block, each wave covers 16x64 -> NACC = 4 (4*4*16 = 256 cols).
    gemm_wmma_f32_lds_kernel<4><<<N / 16, GEMM_BLOCK_THREADS, 0, stream>>>(
        x, W1, h1, N, F_HID, F_IN);

    // self-loop init, edge scatter, relu+bias
    {
        long long total = (long long)N * F_HID;
        self_init_kernel<<<(int)((total + TB - 1) / TB), TB, 0, stream>>>(
            h1, dinv, nullptr, agg1, N, F_HID);
        long long threads = (long long)E * WAVE_SIZE;
        edge_scatter_kernel<<<(int)((threads + TB - 1) / TB), TB, 0, stream>>>(
            h1, src, dst, dinv, agg1, E, F_HID);
        relu_bias_kernel<<<(int)((total + TB - 1) / TB), TB, 0, stream>>>(
            agg1, b1, N, F_HID);
    }

    // --- layer 2: h2 = z1 @ W2  (N x 256) @ (256 x 128) ---------------------
    // 4 waves/block, each wave covers 16x32 -> NACC = 2 (4*2*16 = 128 cols).
    gemm_wmma_f32_lds_kernel<2><<<N / 16, GEMM_BLOCK_THREADS, 0, stream>>>(
        agg1, W2, h2, N, F_OUT, F_HID);

    // self-loop init (+bias b2), edge scatter into d_out
    {
        long long total = (long long)N * F_OUT;
        self_init_kernel<<<(int)((total + TB - 1) / TB), TB, 0, stream>>>(
            h2, dinv, b2, out, N, F_OUT);
        long long threads = (long long)E * WAVE_SIZE;
        edge_scatter_kernel<<<(int)((threads + TB - 1) / TB), TB, 0, stream>>>(
            h2, src, dst, dinv, out, E, F_OUT);
    }
}